// MultiHeadAttentionGraphormerEdge_78692390797923
// MI455X (gfx1250) — compile-verified
//
#include <hip/hip_runtime.h>

typedef float v2f __attribute__((ext_vector_type(2)));
typedef float v8f __attribute__((ext_vector_type(8)));

#define IN_DIM 128
#define HD     128   // H*D
#define NH     8
#define DD     16

// ---------- helpers: monotonic float<->uint mapping for atomic max ----------
__device__ __forceinline__ unsigned fmapu(float x) {
    unsigned u = __float_as_uint(x);
    return (u & 0x80000000u) ? ~u : (u | 0x80000000u);
}
__device__ __forceinline__ float funmap(unsigned m) {
    unsigned u = (m & 0x80000000u) ? (m & 0x7FFFFFFFu) : ~m;
    return __uint_as_float(u);
}

// ---------- Kernel A: fold Aw into Wq/Wk -> Wqk_eff[128][16], biasQK[16] ----
// col c in [0,8): qscore head c ; col c in [8,16): kscore head c-8
__global__ void combine_weights(const float* __restrict__ Wq, const float* __restrict__ bq,
                                const float* __restrict__ Wk, const float* __restrict__ bk,
                                const float* __restrict__ Aw,
                                float* __restrict__ Wqk, float* __restrict__ biasQK) {
    int t = threadIdx.x;
    for (int idx = t; idx < IN_DIM * 16; idx += blockDim.x) {
        int r = idx >> 4, c = idx & 15, h = c & 7;
        const float* W = (c < 8) ? Wq : Wk;
        float acc = 0.f;
        #pragma unroll
        for (int d = 0; d < DD; ++d) acc += W[r * HD + h * DD + d] * Aw[d * NH + h];
        Wqk[idx] = acc;
    }
    if (t < 16) {
        int h = t & 7;
        const float* b = (t < 8) ? bq : bk;
        float acc = 0.f;
        #pragma unroll
        for (int d = 0; d < DD; ++d) acc += b[h * DD + d] * Aw[d * NH + h];
        biasQK[t] = acc;
    }
}

// ---------- Kernel B: S[E][16] = X[E][128] @ Wqk[128][16] + biasQK (WMMA) ---
__global__ __launch_bounds__(256) void score_gemm(const float* __restrict__ X,
                                                  const float* __restrict__ Wqk,
                                                  const float* __restrict__ biasQK,
                                                  float* __restrict__ S, int E) {
    int wave = threadIdx.x >> 5;
    int lane = threadIdx.x & 31;
    int tile = blockIdx.x * 8 + wave;
    int ntiles = E >> 4;
    if (tile >= ntiles) return;               // wave-uniform: EXEC stays all-1s
    int r0   = tile * 16;
    int arow = r0 + (lane & 15);
    int koff = (lane >> 4) << 1;
    int col  = lane & 15;
    v8f c = {};
    for (int k = 0; k < IN_DIM; k += 4) {
        int kb = k + koff;
        v2f a = *(const v2f*)(X + (size_t)arow * IN_DIM + kb);
        v2f b;
        b.x = Wqk[kb * 16 + col];
        b.y = Wqk[(kb + 1) * 16 + col];
        c = __builtin_amdgcn_wmma_f32_16x16x4_f32(false, a, false, b, (short)0, c, false, false);
    }
    float bias = biasQK[col];
    #pragma unroll
    for (int i = 0; i < 8; ++i) {
        int row = r0 + i + ((lane >> 4) << 3);
        S[(size_t)row * 16 + col] = c[i] + bias;
    }
}

// ---------- Kernel C: Vh[E][128] = X[E][128] @ Wv[128][128] + bv (WMMA) ----
// block = 256 thr (8 waves) = 16 rows x 128 cols; A tile staged via LDS
__global__ __launch_bounds__(256) void v_gemm(const float* __restrict__ X,
                                              const float* __restrict__ Wv,
                                              const float* __restrict__ bv,
                                              float* __restrict__ Vh) {
    __shared__ float As[16 * 130];            // stride 130 to avoid bank conflicts
    int t  = threadIdx.x;
    int r0 = blockIdx.x * 16;
    #pragma unroll
    for (int i = 0; i < 8; ++i) {
        int idx = t + i * 256;
        int row = idx >> 7, cx = idx & 127;
        As[row * 130 + cx] = X[(size_t)(r0 + row) * IN_DIM + cx];
    }
    __syncthreads();
    int wave = t >> 5, lane = t & 31;
    int col  = wave * 16 + (lane & 15);
    int koff = (lane >> 4) << 1;
    int arow = lane & 15;
    float bias = bv[col];
    v8f c;
    #pragma unroll
    for (int i = 0; i < 8; ++i) c[i] = bias;
    for (int k = 0; k < IN_DIM; k += 4) {
        int kb = k + koff;
        v2f a = *(const v2f*)(As + arow * 130 + kb);
        v2f b;
        b.x = Wv[kb * HD + col];
        b.y = Wv[(kb + 1) * HD + col];
        c = __builtin_amdgcn_wmma_f32_16x16x4_f32(false, a, false, b, (short)0, c, false, false);
    }
    #pragma unroll
    for (int i = 0; i < 8; ++i) {
        int row = r0 + i + ((lane >> 4) << 3);
        Vh[(size_t)row * HD + col] = c[i];
    }
}

// ---------- Kernel D: init out=0, mxbits=map(-inf), denom=0 ----------------
__global__ void init_bufs(float* __restrict__ out, int nout,
                          unsigned* __restrict__ mxbits, float* __restrict__ denom, int n8) {
    int i = blockIdx.x * blockDim.x + threadIdx.x;
    int stride = gridDim.x * blockDim.x;
    for (int j = i; j < nout; j += stride) out[j] = 0.f;
    for (int j = i; j < n8; j += stride) { mxbits[j] = 0x007FFFFFu; denom[j] = 0.f; }
}

// ---------- Kernel E: segment max over pairs (atomicMax on mapped bits) ----
__global__ void pair_max(const long long* __restrict__ idx, const float* __restrict__ S,
                         unsigned* __restrict__ mxbits, int M) {
    int m = blockIdx.x * blockDim.x + threadIdx.x;
    if (m >= M) return;
    int src = (int)idx[m];
    int dst = (int)idx[M + m];
    const float4* sk = (const float4*)(S + (size_t)src * 16 + 8);
    const float4* sq = (const float4*)(S + (size_t)dst * 16);
    float4 k0 = sk[0], k1 = sk[1], q0 = sq[0], q1 = sq[1];
    float raw[8] = {k0.x + q0.x, k0.y + q0.y, k0.z + q0.z, k0.w + q0.w,
                    k1.x + q1.x, k1.y + q1.y, k1.z + q1.z, k1.w + q1.w};
    unsigned* base = mxbits + (size_t)dst * 8;
    #pragma unroll
    for (int h = 0; h < 8; ++h) atomicMax(base + h, fmapu(raw[h]));
}

// ---------- Kernel F: denom[dst,h] += exp(raw - mx) ------------------------
__global__ void pair_sum(const long long* __restrict__ idx, const float* __restrict__ S,
                         const unsigned* __restrict__ mxbits, float* __restrict__ denom, int M) {
    int m = blockIdx.x * blockDim.x + threadIdx.x;
    if (m >= M) return;
    int src = (int)idx[m];
    int dst = (int)idx[M + m];
    const float4* sk = (const float4*)(S + (size_t)src * 16 + 8);
    const float4* sq = (const float4*)(S + (size_t)dst * 16);
    float4 k0 = sk[0], k1 = sk[1], q0 = sq[0], q1 = sq[1];
    float raw[8] = {k0.x + q0.x, k0.y + q0.y, k0.z + q0.z, k0.w + q0.w,
                    k1.x + q1.x, k1.y + q1.y, k1.z + q1.z, k1.w + q1.w};
    const unsigned* mb = mxbits + (size_t)dst * 8;
    float* dn = denom + (size_t)dst * 8;
    #pragma unroll
    for (int h = 0; h < 8; ++h) atomicAdd(dn + h, __expf(raw[h] - funmap(mb[h])));
}

// ---------- Kernel G: out[dst,h,d] += Vh[src,h,d] * attn -------------------
// one wave per pair; lane owns 4 contiguous (h,d) elems (single head per lane)
__global__ __launch_bounds__(256) void pair_scatter(const long long* __restrict__ idx,
                                                    const float* __restrict__ S,
                                                    const unsigned* __restrict__ mxbits,
                                                    const float* __restrict__ denom,
                                                    const float* __restrict__ Vh,
                                                    float* __restrict__ out, int M) {
    int gid  = blockIdx.x * blockDim.x + threadIdx.x;
    int m    = gid >> 5;
    int lane = gid & 31;
    if (m >= M) return;
    int src = (int)idx[m];
    int dst = (int)idx[M + m];
    int h   = lane >> 2;                       // 4 lanes per head
    float raw  = S[(size_t)src * 16 + 8 + h] + S[(size_t)dst * 16 + h];
    float mx   = funmap(mxbits[(size_t)dst * 8 + h]);
    float attn = __expf(raw - mx) / (denom[(size_t)dst * 8 + h] + 1e-16f);
    float4 v = *(const float4*)(Vh + (size_t)src * HD + lane * 4);
    float* o = out + (size_t)dst * HD + lane * 4;
    atomicAdd(o + 0, v.x * attn);
    atomicAdd(o + 1, v.y * attn);
    atomicAdd(o + 2, v.z * attn);
    atomicAdd(o + 3, v.w * attn);
}

extern "C" void kernel_launch(void* const* d_in, const int* in_sizes, int n_in,
                              void* d_out, int out_size, void* d_ws, size_t ws_size,
                              hipStream_t stream) {
    const float*     edge_attr = (const float*)d_in[0];
    const long long* eeidx     = (const long long*)d_in[1];   // int64 [2, M]
    const float*     Wq        = (const float*)d_in[2];
    const float*     bq        = (const float*)d_in[3];
    const float*     Wk        = (const float*)d_in[4];
    const float*     bk        = (const float*)d_in[5];
    const float*     Wv        = (const float*)d_in[6];
    const float*     bv        = (const float*)d_in[7];
    const float*     Aw        = (const float*)d_in[8];
    float* out = (float*)d_out;

    const int E = in_sizes[0] / IN_DIM;      // 100000
    const int M = in_sizes[1] / 2;           // 1000000

    // workspace layout (floats)
    float*    Vh     = (float*)d_ws;                 // E*128
    float*    S      = Vh + (size_t)E * HD;          // E*16
    unsigned* mxbits = (unsigned*)(S + (size_t)E * 16); // E*8
    float*    denom  = (float*)(mxbits + (size_t)E * 8); // E*8
    float*    Wqk    = denom + (size_t)E * 8;        // 128*16
    float*    biasQK = Wqk + IN_DIM * 16;            // 16

    // A: fold Aw into Q/K weights
    combine_weights<<<1, 256, 0, stream>>>(Wq, bq, Wk, bk, Aw, Wqk, biasQK);

    // B: per-edge q/k scores via WMMA  (ntiles = E/16, 8 tiles per block)
    int ntiles = E >> 4;
    score_gemm<<<(ntiles + 7) / 8, 256, 0, stream>>>(edge_attr, Wqk, biasQK, S, E);

    // C: V projection via WMMA (16 rows per block)
    v_gemm<<<E / 16, 256, 0, stream>>>(edge_attr, Wv, bv, Vh);

    // D: init accumulators
    init_bufs<<<2048, 256, 0, stream>>>(out, E * HD, mxbits, denom, E * NH);

    // E: segment max
    pair_max<<<(M + 255) / 256, 256, 0, stream>>>(eeidx, S, mxbits, M);

    // F: segment sum of exp
    pair_sum<<<(M + 255) / 256, 256, 0, stream>>>(eeidx, S, mxbits, denom, M);

    // G: weighted scatter-add (one wave per pair)
    long long total = (long long)M * 32;
    pair_scatter<<<(unsigned)((total + 255) / 256), 256, 0, stream>>>(eeidx, S, mxbits, denom, Vh, out, M);
}